// SlackADMMRouteB_47433618817422
// MI455X (gfx1250) — compile-verified
//
#include <hip/hip_runtime.h>

// ============================================================================
// ADMM QP solver for m=512, n=1024 (reference: _admm in JAX).
//
// Restructure: grad = c + rho*A^T(Ax - r) == (c - A^T r) + (A^T A) x  (rho=1)
//   Phase 1 (per call): G = A^T A via V_WMMA_F32_16X16X4_F32  (the real GEMM)
//                       At = A^T (for coalesced Ax matvec)
//   Phase 2: 8 persistent blocks, global atomic barrier, 200x(20 inner) loop.
// ============================================================================

#define NN 1024
#define MM 512
#define NB 8
#define N_OUTER 200
#define N_INNER 20
#define X_STEP 1e-3f

typedef __attribute__((ext_vector_type(2))) float v2f;
typedef __attribute__((ext_vector_type(8))) float v8f;

__device__ float    g_G[NN * NN];   // Gram matrix A^T A (symmetric)
__device__ float    g_At[NN * MM];  // At[j*MM + k] = A[k*NN + j]
__device__ float    g_x[NN];
__device__ float    g_s[MM];
__device__ float    g_u[MM];
__device__ unsigned g_bar;          // monotonic grid-barrier counter

// ---------------------------------------------------------------------------
__global__ void init_kernel() {
  int t = threadIdx.x;              // 1024 threads
  g_x[t] = 0.f;
  if (t < MM) { g_s[t] = 0.f; g_u[t] = 0.f; }
  if (t == 0) g_bar = 0u;
}

// ---------------------------------------------------------------------------
__global__ void transpose_kernel(const float* __restrict__ A) {
  int idx = blockIdx.x * blockDim.x + threadIdx.x;   // 0 .. NN*MM-1
  int j = idx >> 9;          // 0..1023  (column of A)
  int k = idx & (MM - 1);    // 0..511   (row of A)
  g_At[j * MM + k] = A[k * NN + j];
}

// ---------------------------------------------------------------------------
// G = A^T A. 64x64 tiles of 16x16; one wave per tile; K-loop of 128 WMMAs.
// f32 WMMA operand layout (ISA 7.12.2):
//   A-op 16x4 : lane<16 -> M=lane, vgpr r holds K=r;  lane>=16 -> K=2+r
//   B-op 4x16 : lane%16 = N,       vgpr r holds K=2*(lane>>4)+r (symmetric)
//   C/D 16x16 : vgpr v, half h -> M = v + 8h, N = lane%16
__global__ void gram_kernel(const float* __restrict__ A) {
  const int lane = threadIdx.x & 31;
  const int wave = threadIdx.x >> 5;
  const int tile = blockIdx.x * 8 + wave;        // 512 blocks * 8 waves = 4096
  const int i0 = (tile >> 6) << 4;
  const int j0 = (tile & 63) << 4;
  const int half = lane >> 4;
  const int l = lane & 15;

  v8f acc = {0.f, 0.f, 0.f, 0.f, 0.f, 0.f, 0.f, 0.f};
  const float* Ai = A + i0 + l;
  const float* Aj = A + j0 + l;
  for (int k0 = 0; k0 < MM; k0 += 4) {
    const int kb = k0 + 2 * half;
    v2f av, bv;
    av[0] = Ai[(kb + 0) * NN];
    av[1] = Ai[(kb + 1) * NN];
    bv[0] = Aj[(kb + 0) * NN];
    bv[1] = Aj[(kb + 1) * NN];
    acc = __builtin_amdgcn_wmma_f32_16x16x4_f32(
        /*neg_a=*/false, av, /*neg_b=*/false, bv,
        /*c_mod=*/(short)0, acc, /*reuse_a=*/false, /*reuse_b=*/false);
  }
#pragma unroll
  for (int v = 0; v < 8; ++v)
    g_G[(i0 + v + 8 * half) * NN + (j0 + l)] = acc[v];
}

// ---------------------------------------------------------------------------
// Grid barrier: monotonic counter (reset by init_kernel each call).
__device__ __forceinline__ void gbar_sync(unsigned* tgt) {
  __syncthreads();
  __threadfence();
  if (threadIdx.x == 0) {
    __hip_atomic_fetch_add(&g_bar, 1u, __ATOMIC_RELEASE,
                           __HIP_MEMORY_SCOPE_AGENT);
    *tgt += NB;
    while (__hip_atomic_load(&g_bar, __ATOMIC_ACQUIRE,
                             __HIP_MEMORY_SCOPE_AGENT) < *tgt)
      __builtin_amdgcn_s_sleep(2);
  }
  __syncthreads();
  __threadfence();   // all threads: invalidate near caches before re-reading
}

// ---------------------------------------------------------------------------
__global__ void __launch_bounds__(1024, 1)
solver_kernel(const float* __restrict__ A, const float* __restrict__ b,
              const float* __restrict__ c, float* __restrict__ out) {
  __shared__ __align__(16) float lx[NN];     // current x (broadcast reads)
  __shared__ float lred[1024];               // partial-sum reduction
  __shared__ __align__(16) float lr[MM];     // r (reused for u at the end)
  __shared__ float lb[MM];
  __shared__ float ld[128];                  // this block's slice of d
  __shared__ float lc[128];

  const int t  = threadIdx.x;
  const int bk = blockIdx.x;
  unsigned tgt = 0;

  if (t < MM)  lb[t] = b[t];
  if (t < 128) lc[t] = c[bk * 128 + t];

  const int o   = bk * 128 + (t & 127);   // owned output index (n-space)
  const int p8  = t >> 7;                 // 0..7
  const int o2  = bk * 64 + (t & 63);     // owned output index (m-space)
  const int p16 = t >> 6;                 // 0..15

  for (int outer = 0; outer < N_OUTER; ++outer) {
    // r = s + b - u
    if (t < MM) lr[t] = g_s[t] + lb[t] - g_u[t];
    __syncthreads();
    // d = c - A^T r   (column access of row-major A: coalesced)
    {
      float acc = 0.f;
      const float* Acol = A + o;
      const float4* lr4 = (const float4*)lr;
      const int kb = p8 * 64;
#pragma unroll 4
      for (int kk = 0; kk < 64; kk += 4) {
        float4 rv = lr4[(kb + kk) >> 2];
        acc += Acol[(kb + kk + 0) * NN] * rv.x +
               Acol[(kb + kk + 1) * NN] * rv.y +
               Acol[(kb + kk + 2) * NN] * rv.z +
               Acol[(kb + kk + 3) * NN] * rv.w;
      }
      lred[t] = acc;
      __syncthreads();
      if (t < 128) {
        float s = 0.f;
#pragma unroll
        for (int p = 0; p < 8; ++p) s += lred[p * 128 + t];
        ld[t] = lc[t] - s;
      }
    }
    // inner projected-gradient loop: x = relu(x - step*(d + G x))
    for (int it = 0; it < N_INNER; ++it) {
      __syncthreads();
      lx[t] = g_x[t];
      __syncthreads();
      float acc = 0.f;
      const float* Gcol = g_G + o;   // G symmetric: G[j][o] == G[o][j]
      const float4* lx4 = (const float4*)lx;
      const int jb = p8 * 128;
#pragma unroll 4
      for (int jj = 0; jj < 128; jj += 4) {
        float4 xv = lx4[(jb + jj) >> 2];
        acc += Gcol[(size_t)(jb + jj + 0) * NN] * xv.x +
               Gcol[(size_t)(jb + jj + 1) * NN] * xv.y +
               Gcol[(size_t)(jb + jj + 2) * NN] * xv.z +
               Gcol[(size_t)(jb + jj + 3) * NN] * xv.w;
      }
      lred[t] = acc;
      __syncthreads();
      if (t < 128) {
        float y = 0.f;
#pragma unroll
        for (int p = 0; p < 8; ++p) y += lred[p * 128 + t];
        float xo = lx[bk * 128 + t];
        float xn = xo - X_STEP * (ld[t] + y);
        g_x[bk * 128 + t] = xn > 0.f ? xn : 0.f;
      }
      gbar_sync(&tgt);
    }
    // Ax via At (coalesced); then s,u updates
    lx[t] = g_x[t];
    __syncthreads();
    {
      float acc = 0.f;
      const float* Atc = g_At + o2;
      const float4* lx4 = (const float4*)lx;
      const int jb = p16 * 64;
#pragma unroll 4
      for (int jj = 0; jj < 64; jj += 4) {
        float4 xv = lx4[(jb + jj) >> 2];
        acc += Atc[(jb + jj + 0) * MM] * xv.x +
               Atc[(jb + jj + 1) * MM] * xv.y +
               Atc[(jb + jj + 2) * MM] * xv.z +
               Atc[(jb + jj + 3) * MM] * xv.w;
      }
      lred[t] = acc;
      __syncthreads();
      if (t < 64) {
        float ax = 0.f;
#pragma unroll
        for (int p = 0; p < 16; ++p) ax += lred[p * 64 + t];
        const int k = bk * 64 + t;
        float uo = g_u[k];
        float v  = ax - lb[k] + uo;
        float sn = v > 0.f ? v : 0.f;
        g_s[k] = sn;
        g_u[k] = uo + ax - sn - lb[k];
      }
    }
    gbar_sync(&tgt);
  }

  // ---- outputs: [x(1024) | s(512) | u(512) | lambda(512) | nu(1024)] ----
  if (t < 128) out[bk * 128 + t] = g_x[bk * 128 + t];
  if (t < 64) {
    const int k = bk * 64 + t;
    out[1024 + k] = g_s[k];
    out[1536 + k] = g_u[k];
    float lam = -g_u[k];                 // rho = 1
    out[2048 + k] = lam > 0.f ? lam : 0.f;
  }
  // nu = relu(c + A^T u)
  if (t < MM) lr[t] = g_u[t];
  __syncthreads();
  {
    float acc = 0.f;
    const float* Acol = A + o;
    const float4* lr4 = (const float4*)lr;
    const int kb = p8 * 64;
#pragma unroll 4
    for (int kk = 0; kk < 64; kk += 4) {
      float4 rv = lr4[(kb + kk) >> 2];
      acc += Acol[(kb + kk + 0) * NN] * rv.x +
             Acol[(kb + kk + 1) * NN] * rv.y +
             Acol[(kb + kk + 2) * NN] * rv.z +
             Acol[(kb + kk + 3) * NN] * rv.w;
    }
    lred[t] = acc;
    __syncthreads();
    if (t < 128) {
      float s = 0.f;
#pragma unroll
      for (int p = 0; p < 8; ++p) s += lred[p * 128 + t];
      float v = lc[t] + s;
      out[2560 + bk * 128 + t] = v > 0.f ? v : 0.f;
    }
  }
}

// ---------------------------------------------------------------------------
extern "C" void kernel_launch(void* const* d_in, const int* in_sizes, int n_in,
                              void* d_out, int out_size, void* d_ws,
                              size_t ws_size, hipStream_t stream) {
  (void)in_sizes; (void)n_in; (void)out_size; (void)d_ws; (void)ws_size;
  const float* A = (const float*)d_in[0];   // 512 x 1024, row-major
  const float* b = (const float*)d_in[1];   // 512
  const float* c = (const float*)d_in[2];   // 1024
  float* out = (float*)d_out;               // 3584 floats

  init_kernel<<<1, 1024, 0, stream>>>();
  transpose_kernel<<<(NN * MM) / 256, 256, 0, stream>>>(A);
  gram_kernel<<<512, 256, 0, stream>>>(A);            // WMMA GEMM: G = A^T A
  solver_kernel<<<NB, 1024, 0, stream>>>(A, b, c, out);
}